// Encoder_2585570312331
// MI455X (gfx1250) — compile-verified
//
#include <hip/hip_runtime.h>

// ---------------------------------------------------------------------------
// Problem constants (from reference):  B=2048, T-1=127, H=128, K1=16, K2=15
// ---------------------------------------------------------------------------
#define B_    2048
#define T1_   127
#define H_    128
#define G_    512                 // 4*H gate width
#define KU_   144                 // H + 16 : augmented state [h | xt_pad16]
#define SWP_  528                 // LDS stride (in float2) per (kgroup,half) row: 528*2 dwords == 32 mod 64
#define SU_   146                 // LDS row stride (floats) for u (conflict-free vs. A-frag pattern)
#define NT1_  (B_ * T1_ * 16)     // X_tilde element count (first output)

typedef float v2f __attribute__((ext_vector_type(2)));
typedef float v8f __attribute__((ext_vector_type(8)));

__device__ __forceinline__ float sigmoid_f(float x) {
    return 1.0f / (1.0f + __expf(-x));
}
__device__ __forceinline__ float tanh_f(float x) {
    // tanh(x) = 1 - 2/(exp(2x)+1); saturates correctly at +-inf
    return 1.0f - 2.0f / (__expf(2.0f * x) + 1.0f);
}

// ---------------------------------------------------------------------------
// Kernel 1: attention weights are TIME-INVARIANT (softmax shift-invariance
// cancels the h/c/ba terms), so precompute a1,a2 per batch row, emit
// X_tilde = a1*x_y (output #1) and xt2 = a2*(X*y_prev) (padded to 16, -> ws).
// ---------------------------------------------------------------------------
__global__ __launch_bounds__(128) void attn_prep_kernel(
    const float* __restrict__ X,        // (B, 127, 15)
    const float* __restrict__ y_prev,   // (B, 127)
    const float* __restrict__ Wa,       // (2H + 127, 1)
    float* __restrict__ Xt_out,         // (B, 127, 16)  -> d_out part 1
    float* __restrict__ xt2_out)        // (B, 127, 16)  -> workspace (padded)
{
    __shared__ float f1[16], f2[16], a1s[16], a2s[16];
    const int b = blockIdx.x, tid = threadIdx.x;

    if (tid < 16) { f1[tid] = 0.f; f2[tid] = 0.f; }
    __syncthreads();

    if (tid < T1_) {
        const float wa = Wa[2 * H_ + tid];          // time-tap of Wa
        const float y  = y_prev[b * T1_ + tid];
        const float* xr = X + ((size_t)b * T1_ + tid) * 15;
        #pragma unroll
        for (int k = 0; k < 15; ++k) {
            const float xv = xr[k];
            atomicAdd(&f1[k], xv * wa);             // ds_add_f32
            atomicAdd(&f2[k], xv * y * wa);
        }
        atomicAdd(&f1[15], y * wa);                 // x_y[...,15] = y_prev
    }
    __syncthreads();

    if (tid == 0) {
        float m1 = -1e30f;
        for (int k = 0; k < 16; ++k) m1 = fmaxf(m1, f1[k]);
        float s1 = 0.f;
        for (int k = 0; k < 16; ++k) { a1s[k] = __expf(f1[k] - m1); s1 += a1s[k]; }
        const float r1 = 1.f / s1;
        for (int k = 0; k < 16; ++k) a1s[k] *= r1;

        float m2 = -1e30f;
        for (int k = 0; k < 15; ++k) m2 = fmaxf(m2, f2[k]);
        float s2 = 0.f;
        for (int k = 0; k < 15; ++k) { a2s[k] = __expf(f2[k] - m2); s2 += a2s[k]; }
        const float r2 = 1.f / s2;
        for (int k = 0; k < 15; ++k) a2s[k] *= r2;
        a2s[15] = 0.f;
    }
    __syncthreads();

    if (tid < T1_) {
        const float y  = y_prev[b * T1_ + tid];
        const float* xr = X + ((size_t)b * T1_ + tid) * 15;
        float* o1 = Xt_out  + ((size_t)b * T1_ + tid) * 16;
        float* o2 = xt2_out + ((size_t)b * T1_ + tid) * 16;
        #pragma unroll
        for (int k = 0; k < 15; ++k) {
            const float xv = xr[k];
            o1[k] = a1s[k] * xv;
            o2[k] = a2s[k] * xv * y;
        }
        o1[15] = a1s[15] * y;
        o2[15] = 0.f;                               // K2 padding
    }
}

// ---------------------------------------------------------------------------
// Kernel 2: persistent LSTM recurrence with f32 WMMA (16x16x4).
//   grid = (128 batch tiles of 16, 2 lstms), block = 256 threads = 8 waves.
//   LDS holds W^T pre-swizzled into B-FRAGMENT order:
//     Wl[(kg*2+half)*SWP_ + n] (as float2) = { W[4kg+half][n], W[4kg+2+half][n] }
//   so each lane's B fragment is ONE aligned ds_load_b64, conflict-free
//   (half-wave offset 528*2 dwords == 32 banks). 304 KB.
//   u state is double-buffered (2 x 16 x 146 f32) -> one barrier per step.
//   Wave w owns hidden chunk [16w,16w+16) for ALL 4 gates, so i/f/g/o are
//   lane-aligned and the cell update is pure per-lane VALU; c stays in VGPRs.
// ---------------------------------------------------------------------------
__global__ __launch_bounds__(256, 1) void lstm_wmma_kernel(
    const float* __restrict__ Wih1, const float* __restrict__ Whh1,
    const float* __restrict__ bih1, const float* __restrict__ bhh1,
    const float* __restrict__ Wih2, const float* __restrict__ Whh2,
    const float* __restrict__ bih2, const float* __restrict__ bhh2,
    const float* __restrict__ xt1,   // (B,127,16) = X_tilde (from d_out)
    const float* __restrict__ xt2,   // (B,127,16) padded (from ws)
    float* __restrict__ Xe)          // (B,127,256) -> d_out part 2
{
    __shared__ float Wl[(KU_ / 4) * 2 * SWP_ * 2];   // 72*528 float2 = 304,128 B
    __shared__ float ub[2 * 16 * SU_];               // double-buffered u: 18,688 B

    const int lstm = blockIdx.y;
    const float* Wih = lstm ? Wih2 : Wih1;
    const float* Whh = lstm ? Whh2 : Whh1;
    const float* bih = lstm ? bih2 : bih1;
    const float* bhh = lstm ? bhh2 : bhh1;
    const float* xt  = lstm ? xt2  : xt1;
    const int    Kin = lstm ? 15 : 16;

    const int tid  = threadIdx.x;
    const int lane = tid & 31;
    const int w    = tid >> 5;            // wave id 0..7 -> hidden chunk
    const int b0   = blockIdx.x * 16;     // batch tile base

    // ---- one-time staging, COALESCED global reads, swizzled LDS writes ----
    // Whh is (512,128) row-major: consecutive tid -> consecutive k (coalesced).
    for (int idx = tid; idx < G_ * H_; idx += 256) {
        const int n  = idx >> 7;          // gate column
        const int k  = idx & 127;         // K index (h part)
        const int kg = k >> 2, rk = k & 3;
        const int half = rk & 1, pos = rk >> 1;     // frag dword select
        Wl[(((kg * 2 + half) * SWP_ + n) << 1) + pos] = Whh[idx];
    }
    // Wih part occupies K rows 128..143 (zero-padded for Kin==15).
    for (int idx = tid; idx < G_ * 16; idx += 256) {
        const int n  = idx >> 4;
        const int kk = idx & 15;
        const int k  = H_ + kk;
        const int kg = k >> 2, rk = k & 3;
        const int half = rk & 1, pos = rk >> 1;
        const float v = (kk < Kin) ? Wih[(size_t)n * Kin + kk] : 0.f;
        Wl[(((kg * 2 + half) * SWP_ + n) << 1) + pos] = v;
    }
    for (int idx = tid; idx < 2 * 16 * SU_; idx += 256) ub[idx] = 0.f;  // h=0 both buffers
    __syncthreads();
    {   // xt for t=0 into buffer 0, rows 128..143
        const int m = tid >> 4, kk = tid & 15;
        ub[m * SU_ + H_ + kk] = xt[(((size_t)(b0 + m)) * T1_) * 16 + kk];
    }
    __syncthreads();

    // per-lane constants
    const int nl   = lane & 15;           // N within tile / M within A
    const int half = lane >> 4;           // half-wave select
    const int hblk = w * 16 + nl;         // hidden index this lane covers
    const float bias0 = bih[0 * H_ + hblk] + bhh[0 * H_ + hblk];  // i
    const float bias1 = bih[1 * H_ + hblk] + bhh[1 * H_ + hblk];  // f
    const float bias2 = bih[2 * H_ + hblk] + bhh[2 * H_ + hblk];  // g
    const float bias3 = bih[3 * H_ + hblk] + bhh[3 * H_ + hblk];  // o
    // B-fragment base for this lane: component pair at ((kg*2+half)*SWP_ + hblk)*2
    const float* wB = &Wl[((half * SWP_ + hblk) << 1)];

    v8f c8 = {0.f, 0.f, 0.f, 0.f, 0.f, 0.f, 0.f, 0.f};   // cell state fragment

    for (int t = 0; t < T1_; ++t) {
        const float* uR = &ub[(t & 1) * 16 * SU_];            // read buffer
        float*       uW = &ub[((t + 1) & 1) * 16 * SU_];      // write buffer
        const float* uA = uR + nl * SU_;                      // A-frag row base

        v8f a0 = {bias0,bias0,bias0,bias0,bias0,bias0,bias0,bias0};
        v8f a1 = {bias1,bias1,bias1,bias1,bias1,bias1,bias1,bias1};
        v8f a2 = {bias2,bias2,bias2,bias2,bias2,bias2,bias2,bias2};
        v8f a3 = {bias3,bias3,bias3,bias3,bias3,bias3,bias3,bias3};

        // gates = u(16x144) @ W(144x512) ; 36 wmma K-steps, fragments are
        // single aligned ds_load_b64 each (A shared by the 4 gate tiles).
        #pragma unroll 6
        for (int k = 0; k < KU_ / 4; ++k) {
            const v2f a = *(const v2f*)(uA + 4 * k + 2 * half);
            const float* wk = wB + k * (4 * SWP_);            // advance 2 half-rows
            const v2f bi = *(const v2f*)(wk + 0 * 256);       // i tile (n + 0*128)
            const v2f bf = *(const v2f*)(wk + 1 * 256);       // f tile
            const v2f bg = *(const v2f*)(wk + 2 * 256);       // g tile
            const v2f bo = *(const v2f*)(wk + 3 * 256);       // o tile
            a0 = __builtin_amdgcn_wmma_f32_16x16x4_f32(false, a, false, bi, (short)0, a0, false, false);
            a1 = __builtin_amdgcn_wmma_f32_16x16x4_f32(false, a, false, bf, (short)0, a1, false, false);
            a2 = __builtin_amdgcn_wmma_f32_16x16x4_f32(false, a, false, bg, (short)0, a2, false, false);
            a3 = __builtin_amdgcn_wmma_f32_16x16x4_f32(false, a, false, bo, (short)0, a3, false, false);
        }

        // elementwise cell update — i/f/g/o are lane-aligned, pure VALU.
        // Writes go to the OTHER u buffer, so no barrier needed before them.
        #pragma unroll
        for (int r = 0; r < 8; ++r) {
            const float ii = sigmoid_f(a0[r]);
            const float ff = sigmoid_f(a1[r]);
            const float gg = tanh_f(a2[r]);
            const float oo = sigmoid_f(a3[r]);
            const float cn = ff * c8[r] + ii * gg;
            c8[r] = cn;
            const float hn = oo * tanh_f(cn);
            const int mrow = r + 8 * half;                 // C/D layout: M = r + 8*(lane/16)
            uW[mrow * SU_ + hblk] = hn;                    // h for next step
            Xe[(((size_t)(b0 + mrow)) * T1_ + t) * 256 + lstm * H_ + hblk] = hn;
        }
        if (t + 1 < T1_) {                // next xt into the write buffer
            const int m = tid >> 4, kk = tid & 15;
            uW[m * SU_ + H_ + kk] = xt[(((size_t)(b0 + m)) * T1_ + (t + 1)) * 16 + kk];
        }
        __syncthreads();                  // single barrier per step (double buffer)
    }
}

// ---------------------------------------------------------------------------
extern "C" void kernel_launch(void* const* d_in, const int* in_sizes, int n_in,
                              void* d_out, int out_size, void* d_ws, size_t ws_size,
                              hipStream_t stream) {
    const float* X      = (const float*)d_in[0];
    const float* y_prev = (const float*)d_in[1];
    const float* Wa     = (const float*)d_in[2];
    // d_in[3] = ba : cancels under softmax shift-invariance
    const float* Wih1 = (const float*)d_in[4];
    const float* Whh1 = (const float*)d_in[5];
    const float* bih1 = (const float*)d_in[6];
    const float* bhh1 = (const float*)d_in[7];
    const float* Wih2 = (const float*)d_in[8];
    const float* Whh2 = (const float*)d_in[9];
    const float* bih2 = (const float*)d_in[10];
    const float* bhh2 = (const float*)d_in[11];

    float* out = (float*)d_out;
    float* Xt  = out;                 // X_tilde   (B,127,16)
    float* Xe  = out + NT1_;          // X_encoded (B,127,256)
    float* xt2 = (float*)d_ws;        // padded LSTM-2 inputs (B,127,16) ~17 MB

    attn_prep_kernel<<<dim3(B_), dim3(128), 0, stream>>>(X, y_prev, Wa, Xt, xt2);

    lstm_wmma_kernel<<<dim3(B_ / 16, 2), dim3(256), 0, stream>>>(
        Wih1, Whh1, bih1, bhh1, Wih2, Whh2, bih2, bhh2, Xt, xt2, Xe);
}